// TASACTDLoss_25915832664440
// MI455X (gfx1250) — compile-verified
//
#include <hip/hip_runtime.h>
#include <hip/hip_bf16.h>

#define TD_GAMMA 0.99f
#define TD_BLOCK 256
#define TD_DEPTH 8   // pipeline stages; 7 async loads/stage -> 56 <= 63 (6-bit ASYNCcnt)

#if defined(__AMDGCN__) &&                                              \
    __has_builtin(__builtin_amdgcn_global_load_async_to_lds_b32) &&     \
    __has_builtin(__builtin_amdgcn_s_wait_asynccnt)
#define TD_ASYNC 1
#else
#define TD_ASYNC 0
#endif

__global__ __launch_bounds__(TD_BLOCK) void td_loss_kernel(
    const float* __restrict__ reward,
    const float* __restrict__ discount,
    const float* __restrict__ value,
    const float* __restrict__ target,
    const int*   __restrict__ step_type,
    const int*   __restrict__ rollout_b,
    const int*   __restrict__ train_b,
    float* __restrict__ loss,
    int T, int B)
{
  const int tid = threadIdx.x;
  long long b = (long long)blockIdx.x * TD_BLOCK + tid;
  if (b >= B) b = B - 1;  // keep all lanes active with valid addresses

  float acc = 0.0f, prev_d = 0.0f, prev_r = 0.0f;

#if TD_ASYNC
  // [stage][array][lane]; lane-stride 4B -> bank-conflict-free. 8*7*256*4 = 56 KB.
  __shared__ unsigned smem[TD_DEPTH][7][TD_BLOCK];

  auto issue = [&](int t, int slot) {
    const long long idx = (long long)t * B + b;
#define TD_ASY(k, p)                                                         \
    __builtin_amdgcn_global_load_async_to_lds_b32(                           \
        (__attribute__((address_space(1))) int*)((p) + idx),                 \
        (__attribute__((address_space(3))) int*)(&smem[slot][k][tid]),       \
        0, 0)
    TD_ASY(0, reward);
    TD_ASY(1, discount);
    TD_ASY(2, value);
    TD_ASY(3, target);
    TD_ASY(4, step_type);
    TD_ASY(5, rollout_b);
    TD_ASY(6, train_b);
#undef TD_ASY
  };

  auto consume = [&](int t, int slot) {
    const float r  = __uint_as_float(smem[slot][0][tid]);
    const float d  = __uint_as_float(smem[slot][1][tid]) * TD_GAMMA;
    const float v  = __uint_as_float(smem[slot][2][tid]);
    const float tv = __uint_as_float(smem[slot][3][tid]);
    const int   st = (int)smem[slot][4][tid];
    const int   ro = (int)smem[slot][5][tid];
    const int   tr = (int)smem[slot][6][tid];
    const long long idx = (long long)t * B + b;
    if (t == T - 1) {
      acc = tv;
      loss[idx] = 0.0f;
    } else {
      const float ret = fmaf(acc, prev_d, prev_r);   // acc*discounts[t+1] + reward[t+1]
      bool il = (st == 2) || (ro != 0) || (tr != 0);
      if (t == 0) il = (st == 2);                    // b mask forced false at t==0
      acc = il ? tv : ret;
      const float diff = ret - v;
      loss[idx] = diff * diff;
    }
    prev_d = d;
    prev_r = r;
  };

  // Prologue: stage rows T-1 .. T-DEPTH.
  for (int k = 0; k < TD_DEPTH; ++k)
    if (T - 1 - k >= 0) issue(T - 1 - k, k);

  // Steady state: wait for the oldest stage (async loads retire in order),
  // consume it, refill the slot with row t-DEPTH.
  for (int t = T - 1; t >= TD_DEPTH - 1; --t) {
    __builtin_amdgcn_s_wait_asynccnt(7 * (TD_DEPTH - 1));
    const int slot = (T - 1 - t) & (TD_DEPTH - 1);
    consume(t, slot);
    const int tn = t - TD_DEPTH;
    if (tn >= 0) issue(tn, slot);
  }

  // Tail: everything already issued; drain and consume from LDS.
  __builtin_amdgcn_s_wait_asynccnt(0);
  for (int t = (T < TD_DEPTH ? T : TD_DEPTH) - 2; t >= 0; --t)
    consume(t, (T - 1 - t) & (TD_DEPTH - 1));

#else  // Fallback: direct loads + global_prefetch_b8 lookahead.
  const int PF = 12;
  for (int t = T - 1; t >= 0; --t) {
    const long long idx = (long long)t * B + b;
    if (t >= PF) {
      const long long pidx = idx - (long long)PF * B;
      __builtin_prefetch(reward + pidx, 0, 0);
      __builtin_prefetch(discount + pidx, 0, 0);
      __builtin_prefetch(value + pidx, 0, 0);
      __builtin_prefetch(target + pidx, 0, 0);
      __builtin_prefetch(step_type + pidx, 0, 0);
      __builtin_prefetch(rollout_b + pidx, 0, 0);
      __builtin_prefetch(train_b + pidx, 0, 0);
    }
    const float r  = reward[idx];
    const float d  = discount[idx] * TD_GAMMA;
    const float v  = value[idx];
    const float tv = target[idx];
    const int   st = step_type[idx];
    const int   ro = rollout_b[idx];
    const int   tr = train_b[idx];
    if (t == T - 1) {
      acc = tv;
      loss[idx] = 0.0f;
    } else {
      const float ret = fmaf(acc, prev_d, prev_r);
      bool il = (st == 2) || (ro != 0) || (tr != 0);
      if (t == 0) il = (st == 2);
      acc = il ? tv : ret;
      const float diff = ret - v;
      loss[idx] = diff * diff;
    }
    prev_d = d;
    prev_r = r;
  }
#endif
}

extern "C" void kernel_launch(void* const* d_in, const int* in_sizes, int n_in,
                              void* d_out, int out_size, void* d_ws, size_t ws_size,
                              hipStream_t stream) {
  const float* reward   = (const float*)d_in[0];
  const float* discount = (const float*)d_in[1];
  const float* value    = (const float*)d_in[2];
  const float* target   = (const float*)d_in[3];
  const int*   step_t   = (const int*)d_in[4];
  const int*   roll_b   = (const int*)d_in[5];
  const int*   train_b  = (const int*)d_in[6];
  float* loss = (float*)d_out;

  const int T = 256;
  const int B = in_sizes[0] / T;   // 65536

  dim3 grid((B + TD_BLOCK - 1) / TD_BLOCK);
  dim3 block(TD_BLOCK);
  hipLaunchKernelGGL(td_loss_kernel, grid, block, 0, stream,
                     reward, discount, value, target, step_t, roll_b, train_b,
                     loss, T, B);
}